// QuantizedEmbedding_28458453303848
// MI455X (gfx1250) — compile-verified
//
#include <hip/hip_runtime.h>
#include <stdint.h>

// ---------------------------------------------------------------------------
// QuantizedEmbedding gather+dequant for MI455X (gfx1250).
//   out[t, :] = (float)weight[input[t], :] * weight_scale[input[t]]
// Memory-bound: ~512 MB traffic -> ~22us at 23.3 TB/s. No matrix math, so no
// WMMA; instead we use the CDNA5 async global->LDS copy engine (ASYNCcnt) to
// double-buffer 2KB row chunks per wave, hiding random-gather HBM latency
// without spending VGPRs on in-flight load data.
// ---------------------------------------------------------------------------

typedef __attribute__((ext_vector_type(4))) float v4f;
typedef __attribute__((ext_vector_type(4))) int   v4i;

#define WAVES_PER_BLOCK 8
#define BLOCK_THREADS   (WAVES_PER_BLOCK * 32)
#define CHUNK_INTS      512                    // 2 KB chunk
#define CHUNK_BYTES     (CHUNK_INTS * 4)

// Issue one 2KB chunk: 4 async B128 ops (each moves 16B/lane * 32 lanes = 512B).
// VDST operand = LDS byte offset (low 32 bits of the flat shared pointer),
// VADDR = 64-bit global address, saddr = off.
__device__ __forceinline__ void issue_chunk_async(const int* gsrc,
                                                  unsigned lds_dst,
                                                  int lane) {
    const char* g = (const char*)gsrc + lane * 16;
    unsigned     l = lds_dst + lane * 16;
#pragma unroll
    for (int j = 0; j < 4; ++j) {
        asm volatile("global_load_async_to_lds_b128 %0, %1, off"
                     :: "v"(l), "v"(g)
                     : "memory");
        g += 512;
        l += 512;
    }
}

__global__ __launch_bounds__(BLOCK_THREADS)
void dequant_gather_async(const int* __restrict__ tokens,
                          const int* __restrict__ weight,
                          const float* __restrict__ wscale,
                          float* __restrict__ out,
                          int nTokens, int D, int cprShift) {
    // 2 buffers x 8 waves x 2KB = 32 KB static LDS.
    __shared__ int smem[2 * WAVES_PER_BLOCK * CHUNK_INTS];

    const int lane = threadIdx.x & 31;
    const int wid  = threadIdx.x >> 5;
    const int gw   = blockIdx.x * WAVES_PER_BLOCK + wid;     // global wave id
    const int nw   = gridDim.x * WAVES_PER_BLOCK;            // total waves
    const int cpr  = 1 << cprShift;                          // chunks per row
    const long totalChunks = (long)nTokens << cprShift;

    int* bufs = &smem[wid * (2 * CHUNK_INTS)];
    // Low 32 bits of the flat shared-space address == LDS byte offset
    // (flat LDS aperture mapping, ISA 10.2).
    const unsigned ldsBase = (unsigned)(uintptr_t)bufs;

    long c = gw;
    if (c >= totalChunks) return;

    // Prologue: start streaming the first chunk.
    {
        const int row  = (int)(c >> cprShift);
        const int part = (int)(c & (cpr - 1));
        const int tok  = __builtin_amdgcn_readfirstlane(tokens[row]);
        issue_chunk_async(weight + (size_t)tok * D + part * CHUNK_INTS,
                          ldsBase, lane);
    }

    int cur = 0;
    for (; c < totalChunks; c += nw) {
        const long cn = c + nw;
        if (cn < totalChunks) {
            // Prefetch next chunk into the other buffer, then wait until only
            // those 4 ops are outstanding -> current buffer is complete
            // (async ops complete in order per wave).
            const int rowN  = (int)(cn >> cprShift);
            const int partN = (int)(cn & (cpr - 1));
            const int tokN  = __builtin_amdgcn_readfirstlane(tokens[rowN]);
            issue_chunk_async(weight + (size_t)tokN * D + partN * CHUNK_INTS,
                              ldsBase + (unsigned)((cur ^ 1) * CHUNK_BYTES),
                              lane);
            asm volatile("s_wait_asynccnt 4" ::: "memory");
        } else {
            asm volatile("s_wait_asynccnt 0" ::: "memory");
        }

        // Dequantize current chunk out of LDS and stream to HBM.
        const int   row  = (int)(c >> cprShift);
        const int   part = (int)(c & (cpr - 1));
        const int   tok  = __builtin_amdgcn_readfirstlane(tokens[row]);
        const float s    = wscale[tok];

        const v4i* src = (const v4i*)(bufs + cur * CHUNK_INTS);
        v4f* dst = (v4f*)(out + (size_t)row * D + part * CHUNK_INTS);
#pragma unroll
        for (int j = 0; j < 4; ++j) {
            v4i w = src[j * 32 + lane];
            v4f f;
            f.x = (float)w.x * s;
            f.y = (float)w.y * s;
            f.z = (float)w.z * s;
            f.w = (float)w.w * s;
            // Output is write-once: non-temporal, don't pollute L2.
            __builtin_nontemporal_store(f, &dst[j * 32 + lane]);
        }
        cur ^= 1;
    }
}

// Fallback for shapes where D is not a power-of-two multiple of 512.
__global__ void dequant_gather_simple(const int* __restrict__ tokens,
                                      const int* __restrict__ weight,
                                      const float* __restrict__ wscale,
                                      float* __restrict__ out,
                                      int D, long total) {
    long i = (long)blockIdx.x * blockDim.x + threadIdx.x;
    if (i >= total) return;
    long row = i / D;
    long col = i - row * D;
    int  tok = tokens[row];
    float s  = wscale[tok];
    out[i]   = (float)weight[(size_t)tok * D + col] * s;
}

extern "C" void kernel_launch(void* const* d_in, const int* in_sizes, int n_in,
                              void* d_out, int out_size, void* d_ws, size_t ws_size,
                              hipStream_t stream) {
    const int*   tokens = (const int*)d_in[0];    // [8,4096] token ids
    const int*   weight = (const int*)d_in[1];    // [V, D] int8-as-int32
    const float* wscale = (const float*)d_in[2];  // [V]
    float*       out    = (float*)d_out;          // [8,4096,D] f32

    const int  N = in_sizes[0];
    const long weightElems = (long)in_sizes[1];
    const int  V = in_sizes[2];
    const int  D = (int)(weightElems / V);

    const int cpr = D / CHUNK_INTS;
    const bool fastOK = (cpr > 0) && ((cpr & (cpr - 1)) == 0) &&
                        (cpr * CHUNK_INTS == D);

    if (fastOK) {
        const int cprShift = __builtin_ctz((unsigned)cpr);
        const int blocks = 1024;   // 8192 waves; 131072 chunks -> 16 chunks/wave
        dequant_gather_async<<<blocks, BLOCK_THREADS, 0, stream>>>(
            tokens, weight, wscale, out, N, D, cprShift);
    } else {
        const long total = (long)N * D;
        const int threads = 256;
        const long blocks = (total + threads - 1) / threads;
        dequant_gather_simple<<<(int)blocks, threads, 0, stream>>>(
            tokens, weight, wscale, out, D, total);
    }
}